// TopKRouter_61942018343436
// MI455X (gfx1250) — compile-verified
//
#include <hip/hip_runtime.h>
#include <math.h>

// MoE Top-K sigmoid router for MI455X (gfx1250, wave32).
// logits = x[T,4096] @ W[64,4096]^T via V_WMMA_F32_16X16X4_F32 (fp32-exact),
// 4-way split-K per block, LDS ds_add_f32 reduction, fused sigmoid/top-8.

typedef __attribute__((ext_vector_type(2))) float v2f;
typedef __attribute__((ext_vector_type(8))) float v8f;

#define HDIM 4096
#define NEXP 64
#define TOPK 8
#define KSLICE (HDIM / 4)          // 1024 K per wave
#define KSTEPS (KSLICE / 4)        // 256 WMMA K-steps per wave

__global__ __launch_bounds__(128) void topk_router_kernel(
    const float* __restrict__ x,
    const float* __restrict__ w,
    const float* __restrict__ bias,
    float* __restrict__ out,
    int T)
{
    __shared__ float s_scores[16][NEXP];   // 4 KB logits tile for this block
    __shared__ float s_bias[NEXP];

    const int tid  = threadIdx.x;
    const int wave = tid >> 5;             // 0..3 : K-slice owner
    const int lane = tid & 31;
    const int half = lane >> 4;            // selects K=0,1 vs K=2,3 within a WMMA step
    const int lc   = lane & 15;            // A row (token) / B column (expert) within tile
    const int tok0 = blockIdx.x * 16;

    // Zero the accumulation tile, stage expert bias in LDS.
    for (int i = tid; i < 16 * NEXP; i += 128)
        ((float*)s_scores)[i] = 0.0f;
    if (tid < NEXP)
        s_bias[tid] = bias[tid];
    __syncthreads();

    // ---- GEMM phase: D(16x16 f32) += A(16x4 f32) x B(4x16 f32) ----
    // A: lane lc holds row (tok0+lc); VGPR0/1 = K = kbase + 2*half + {0,1}
    // B: lane lc holds col (expert); same K pattern -> contiguous float2 loads.
    // Fold the per-lane K start into the base pointers so the loop trip count
    // is a uniform compile-time constant (scalar branch, unrollable).
    const int kbeg = wave * KSLICE + 2 * half;
    const float* xp = x + (size_t)(tok0 + lc) * HDIM + kbeg;
    const float* p0 = w + (size_t)lc * HDIM + kbeg;
    const float* p1 = p0 + (size_t)16 * HDIM;
    const float* p2 = p0 + (size_t)32 * HDIM;
    const float* p3 = p0 + (size_t)48 * HDIM;

    v8f acc0 = {}; v8f acc1 = {}; v8f acc2 = {}; v8f acc3 = {};

#pragma unroll 4
    for (int it = 0; it < KSTEPS; ++it) {
        const int k = it * 4;
        v2f a  = *(const v2f*)(xp + k);
        v2f b0 = *(const v2f*)(p0 + k);
        v2f b1 = *(const v2f*)(p1 + k);
        v2f b2 = *(const v2f*)(p2 + k);
        v2f b3 = *(const v2f*)(p3 + k);
        acc0 = __builtin_amdgcn_wmma_f32_16x16x4_f32(false, a, false, b0, (short)0, acc0, false, false);
        acc1 = __builtin_amdgcn_wmma_f32_16x16x4_f32(false, a, false, b1, (short)0, acc1, false, false);
        acc2 = __builtin_amdgcn_wmma_f32_16x16x4_f32(false, a, false, b2, (short)0, acc2, false, false);
        acc3 = __builtin_amdgcn_wmma_f32_16x16x4_f32(false, a, false, b3, (short)0, acc3, false, false);
    }

    // ---- Split-K reduction into LDS with ds_add_f32 atomics ----
    // C/D layout: VGPR i of acc = row (8*half + i), column = lc (per expert tile).
#pragma unroll
    for (int i = 0; i < 8; ++i) {
        const int tr = half * 8 + i;
        atomicAdd(&s_scores[tr][lc],      acc0[i]);
        atomicAdd(&s_scores[tr][lc + 16], acc1[i]);
        atomicAdd(&s_scores[tr][lc + 32], acc2[i]);
        atomicAdd(&s_scores[tr][lc + 48], acc3[i]);
    }
    __syncthreads();

    // ---- Routing phase: one lane per token ----
    if (tid < 16) {
        float* row = s_scores[tid];
        // sigmoid scores in-place
        for (int e = 0; e < NEXP; ++e)
            row[e] = 1.0f / (1.0f + __expf(-row[e]));

        float sel[TOPK];
        int   idx[TOPK];
        float sum = 0.0f;
#pragma unroll
        for (int p = 0; p < TOPK; ++p) {
            float best = -1e30f;
            int   be   = 0;
            for (int e = 0; e < NEXP; ++e) {
                const float v = row[e] + s_bias[e];   // bias only affects selection
                if (v > best) { best = v; be = e; }   // strict > == lowest-index tie-break
            }
            sel[p] = row[be];
            idx[p] = be;
            row[be] = -1e30f;                          // knock out for next pass
            sum += sel[p];
        }
        const float inv = 1.0f / (sum + 1e-20f);
        const size_t gt = (size_t)(tok0 + tid);
#pragma unroll
        for (int p = 0; p < TOPK; ++p) {
            out[gt * TOPK + p]                    = sel[p] * inv;   // probs
            out[(size_t)T * TOPK + gt * TOPK + p] = (float)idx[p];  // indices
        }
    }
}

extern "C" void kernel_launch(void* const* d_in, const int* in_sizes, int n_in,
                              void* d_out, int out_size, void* d_ws, size_t ws_size,
                              hipStream_t stream) {
    (void)n_in; (void)out_size; (void)d_ws; (void)ws_size;
    const float* x    = (const float*)d_in[0];
    const float* w    = (const float*)d_in[1];
    const float* bias = (const float*)d_in[2];
    float* out = (float*)d_out;

    const int T = in_sizes[0] / HDIM;      // 16384
    const int grid = T / 16;               // 16 tokens per block
    topk_router_kernel<<<grid, 128, 0, stream>>>(x, w, bias, out, T);
}